// MFCC_2619930051002
// MI455X (gfx1250) — compile-verified
//
#include <hip/hip_runtime.h>
#include <math.h>

// ---------------------------------------------------------------------------
// MFCC pipeline for MI455X (gfx1250, wave32, WMMA).
// Spectrum = DFT-as-GEMM using V_WMMA_F32_16X16X4_F32.
// Round 2: branch-free inner loop (clamped row + zero multiplier), two
// N-tiles per job (A reuse, 4 independent WMMA chains per wave).
// ---------------------------------------------------------------------------

typedef __attribute__((ext_vector_type(2))) float v2f;
typedef __attribute__((ext_vector_type(8))) float v8f;

#define NSAMP   16000
#define WIN     640
#define HOP     320
#define NFRM    49          // 1 + (16000-640)/320
#define NBIN    513
#define NBIN_P  544         // 34 tiles of 16 (pad so N-tiles always pair up)
#define NMEL    40
#define NCEP    13
#define CWP     132         // padded LDS chunk row pitch (128 bins + 4)

static __device__ inline v8f wmma_f32_16x16x4(v2f a, v2f b, v8f c) {
    // (neg_a, A, neg_b, B, c_mod, C, reuse_a, reuse_b)
    return __builtin_amdgcn_wmma_f32_16x16x4_f32(
        false, a, false, b, (short)0, c, false, false);
}

// ---------------------------------------------------------------------------
// Transposed DFT bases: basisT[bin][n] = cos/sin(2*pi*bin*n/1024), exact
// mod-1024 twiddle reduction. Rows 513..543 are padding (finite, unused).
// ---------------------------------------------------------------------------
__global__ __launch_bounds__(256) void mfcc_basis_kernel(float* __restrict__ bc,
                                                         float* __restrict__ bs) {
    int idx = blockIdx.x * 256 + threadIdx.x;
    if (idx >= NBIN_P * WIN) return;
    int j = idx / WIN;
    int n = idx - j * WIN;
    int m = (j * n) & 1023;                       // exact twiddle reduction
    float ang = (float)m * (6.283185307179586f / 1024.0f);
    float s, c;
    sincosf(ang, &s, &c);
    bc[idx] = c;
    bs[idx] = s;
}

// ---------------------------------------------------------------------------
// One workgroup (256 threads = 8 wave32) per batch element.
// ---------------------------------------------------------------------------
__global__ __launch_bounds__(256) void mfcc_main_kernel(
    const float* __restrict__ wav,      // [1024][16000]
    const float* __restrict__ melfb,    // [513][40]
    const float* __restrict__ basC,     // [544][640]
    const float* __restrict__ basS,     // [544][640]
    float* __restrict__ out)            // [1024][49][39]
{
    __shared__ float s_win[WIN];            //  2.5 KB
    __shared__ float s_spec[64 * CWP];      // 33.0 KB  (64 frame rows x 132)
    __shared__ float s_mel[64 * NMEL];      // 10.0 KB
    __shared__ float s_dct[NMEL * NCEP];    //  2.0 KB
    __shared__ float s_mf[64 * NCEP];       //  3.3 KB
    __shared__ float s_d1[64 * NCEP];       //  3.3 KB

    const int b    = blockIdx.x;
    const int tid  = threadIdx.x;
    const int wv   = tid >> 5;
    const int lane = tid & 31;
    const int laneM  = lane & 15;
    const int laneHi = lane >> 4;           // 0 or 1
    const int nBase  = 2 * laneHi;          // K offset inside a 4-step

    // ---- phase 0: window + DCT matrix into LDS --------------------------
    for (int n = tid; n < WIN; n += 256)
        s_win[n] = 0.5f - 0.5f * __cosf(6.283185307179586f * (float)n / (float)WIN);
    for (int p = tid; p < NMEL * NCEP; p += 256) {
        int m = p / NCEP, k = p - m * NCEP;
        s_dct[p] = 2.0f * __cosf(3.14159265358979f * (2.0f * m + 1.0f) * (float)k
                                 / (2.0f * (float)NMEL));
    }
    __syncthreads();

    // per-thread mel accumulators: thread owns pairs p = tid + i*256,
    // frame = p/40, mel = p%40  (64*40 = 2560 pairs total)
    float macc[10];
#pragma unroll
    for (int i = 0; i < 10; ++i) macc[i] = 0.0f;

    // ---- spectrum (WMMA DFT) + mel projection, 5 bin-chunks -------------
    for (int c = 0; c < 5; ++c) {
        const int pairs = (c < 4) ? 4 : 1;      // N-tile pairs in this chunk
        const int jobs  = 4 * pairs;            // 4 M-tiles x pairs

        for (int job = wv; job < jobs; job += 8) {
            const int tm   = job / pairs;
            const int pn   = job - tm * pairs;
            const int tnl  = pn * 2;                    // local N tile (even)
            const int bin0 = (c * 8 + tnl) * 16;

            // A side: clamped frame row + zero multiplier (no divergence)
            const int  rA     = tm * 16 + laneM;
            const float fscale = (rA < NFRM) ? 1.0f : 0.0f;
            const int  rC     = (rA < NFRM) ? rA : (NFRM - 1);
            const float* wrow = wav + (size_t)b * NSAMP + (size_t)rC * HOP;

            const float2* bcA = (const float2*)(basC + (size_t)(bin0 + laneM) * WIN);
            const float2* bsA = (const float2*)(basS + (size_t)(bin0 + laneM) * WIN);
            const float2* bcB = (const float2*)(basC + (size_t)(bin0 + 16 + laneM) * WIN);
            const float2* bsB = (const float2*)(basS + (size_t)(bin0 + 16 + laneM) * WIN);

            v8f accR0 = {0.f,0.f,0.f,0.f,0.f,0.f,0.f,0.f};
            v8f accI0 = {0.f,0.f,0.f,0.f,0.f,0.f,0.f,0.f};
            v8f accR1 = {0.f,0.f,0.f,0.f,0.f,0.f,0.f,0.f};
            v8f accI1 = {0.f,0.f,0.f,0.f,0.f,0.f,0.f,0.f};

#pragma unroll 4
            for (int kk = 0; kk < WIN; kk += 4) {
                const int n0 = kk + nBase;          // even -> 8B aligned
                float2 xv = *(const float2*)(wrow + n0);
                float2 wn = *(const float2*)(&s_win[n0]);
                v2f a;
                a.x = xv.x * wn.x * fscale;
                a.y = xv.y * wn.y * fscale;

                float2 c0 = bcA[n0 >> 1];
                float2 s0 = bsA[n0 >> 1];
                float2 c1 = bcB[n0 >> 1];
                float2 s1 = bsB[n0 >> 1];
                v2f bR0; bR0.x = c0.x; bR0.y = c0.y;
                v2f bI0; bI0.x = s0.x; bI0.y = s0.y;
                v2f bR1; bR1.x = c1.x; bR1.y = c1.y;
                v2f bI1; bI1.x = s1.x; bI1.y = s1.y;

                accR0 = wmma_f32_16x16x4(a, bR0, accR0);
                accI0 = wmma_f32_16x16x4(a, bI0, accI0);
                accR1 = wmma_f32_16x16x4(a, bR1, accR1);
                accI1 = wmma_f32_16x16x4(a, bI1, accI1);
            }

            // C/D layout: VGPR v -> row (v + 8*laneHi), col laneM
            const int rBase = tm * 16 + 8 * laneHi;
#pragma unroll
            for (int v = 0; v < 8; ++v) {
                float re0 = accR0[v], im0 = accI0[v];
                float re1 = accR1[v], im1 = accI1[v];
                s_spec[(rBase + v) * CWP + tnl * 16 + laneM] =
                    sqrtf(re0 * re0 + im0 * im0);
                s_spec[(rBase + v) * CWP + (tnl + 1) * 16 + laneM] =
                    sqrtf(re1 * re1 + im1 * im1);
            }
        }
        __syncthreads();

        // mel projection for this bin chunk (register accumulation)
        const int bg0 = c * 128;
        const int nb  = (NBIN - bg0 < 128) ? (NBIN - bg0) : 128;
#pragma unroll
        for (int i = 0; i < 10; ++i) {
            int p = tid + i * 256;
            int f = p / NMEL, m = p - f * NMEL;
            if (f < NFRM) {
                float s = 0.0f;
                for (int bl = 0; bl < nb; ++bl)
                    s = fmaf(s_spec[f * CWP + bl],
                             melfb[(size_t)(bg0 + bl) * NMEL + m], s);
                macc[i] += s;
            }
        }
        __syncthreads();   // before next chunk overwrites s_spec
    }

    // ---- log-mel --------------------------------------------------------
#pragma unroll
    for (int i = 0; i < 10; ++i) {
        int p = tid + i * 256;
        int f = p / NMEL, m = p - f * NMEL;
        if (f < NFRM) s_mel[f * NMEL + m] = logf(macc[i] + 1e-6f);
    }
    __syncthreads();

    // ---- DCT + lifter ---------------------------------------------------
    const float scale = 0.1118033988749895f;    // 1/sqrt(2*NMEL) = 1/sqrt(80)
    for (int p = tid; p < NFRM * NCEP; p += 256) {
        int f = p / NCEP, k = p - f * NCEP;
        float s = 0.0f;
        for (int m = 0; m < NMEL; ++m)
            s = fmaf(s_mel[f * NMEL + m], s_dct[m * NCEP + k], s);
        float lift = 1.0f + 11.0f * __sinf(3.14159265358979f * (float)k / 22.0f);
        s_mf[f * NCEP + k] = s * scale * lift;
    }
    __syncthreads();

    // ---- delta (edge-replicated, N=2, denom=10) -------------------------
    for (int p = tid; p < NFRM * NCEP; p += 256) {
        int f = p / NCEP, k = p - f * NCEP;
        float acc = 0.0f;
#pragma unroll
        for (int d = 1; d <= 2; ++d) {
            int fp = f + d; if (fp > NFRM - 1) fp = NFRM - 1;
            int fm = f - d; if (fm < 0) fm = 0;
            acc += (float)d * (s_mf[fp * NCEP + k] - s_mf[fm * NCEP + k]);
        }
        s_d1[f * NCEP + k] = acc * 0.1f;
    }
    __syncthreads();

    // ---- delta-delta + write out [b][f][0..38] --------------------------
    float* ob = out + (size_t)b * NFRM * 39;
    for (int p = tid; p < NFRM * NCEP; p += 256) {
        int f = p / NCEP, k = p - f * NCEP;
        float acc = 0.0f;
#pragma unroll
        for (int d = 1; d <= 2; ++d) {
            int fp = f + d; if (fp > NFRM - 1) fp = NFRM - 1;
            int fm = f - d; if (fm < 0) fm = 0;
            acc += (float)d * (s_d1[fp * NCEP + k] - s_d1[fm * NCEP + k]);
        }
        ob[f * 39 + k]          = s_mf[f * NCEP + k];
        ob[f * 39 + 13 + k]     = s_d1[f * NCEP + k];
        ob[f * 39 + 26 + k]     = acc * 0.1f;
    }
}

// ---------------------------------------------------------------------------
extern "C" void kernel_launch(void* const* d_in, const int* in_sizes, int n_in,
                              void* d_out, int out_size, void* d_ws, size_t ws_size,
                              hipStream_t stream) {
    const float* wav   = (const float*)d_in[0];   // (1024, 16000) f32
    const float* melfb = (const float*)d_in[1];   // (513, 40) f32
    float* basC = (float*)d_ws;                   // 544*640 floats
    float* basS = basC + NBIN_P * WIN;            // 544*640 floats
    float* out  = (float*)d_out;                  // (1024, 49, 39, 1) f32

    const int basisElems = NBIN_P * WIN;
    mfcc_basis_kernel<<<(basisElems + 255) / 256, 256, 0, stream>>>(basC, basS);
    mfcc_main_kernel<<<1024, 256, 0, stream>>>(wav, melfb, basC, basS, out);
}